// DemoHandler_86303072846400
// MI455X (gfx1250) — compile-verified
//
#include <hip/hip_runtime.h>
#include <math.h>

// ---------------------------------------------------------------------------
// Hyperbolic graph attention (Poincare ball, c=1) for MI455X / gfx1250.
//   pass A: per-edge log-map + WMMA MLP score -> atomic segment max
//   pass B: recompute score -> exp(score-max), atomic segment sums
//   pass C: per-node mixer + exp-map update
// ---------------------------------------------------------------------------

static constexpr int V    = 2097152;
static constexpr int E    = 8388608;
static constexpr float EPSF      = 1e-15f;
static constexpr float MAX_ATANH = 0.99999f;          // 1.0 - 1e-5

typedef __attribute__((ext_vector_type(2))) float v2f;
typedef __attribute__((ext_vector_type(8))) float v8f;

// monotonic float -> uint key so atomicMax(uint) == float max
__device__ __forceinline__ unsigned fkey(float f) {
    unsigned u = __float_as_uint(f);
    return (u & 0x80000000u) ? ~u : (u | 0x80000000u);
}
__device__ __forceinline__ float fdec(unsigned k) {
    return __uint_as_float((k & 0x80000000u) ? (k & 0x7fffffffu) : ~k);
}

__device__ __forceinline__ float gelu_exact(float h) {
    return 0.5f * h * (1.0f + erff(h * 0.70710678118654752f));
}

// Whole-wave helper: lane L owns edge L (0..31). Two 16-edge tiles, each a
// V_WMMA_F32_16X16X4_F32: A = [16 edges x K(2, padded to 4)], B = W1 (4x16,
// rows 2..3 zero), C = bias broadcast.  Second layer (16->1) = per-lane
// multiply by W2[n] + 16-wide butterfly reduction, then realign to edge lanes.
__device__ __forceinline__ float wave_mlp_score(float vx, float vy,
                                                float w1_0, float w1_1,
                                                float bias, float w2v) {
    const int lane = threadIdx.x & 31;
    const bool lo  = lane < 16;

    v2f B; B.x = lo ? w1_0 : 0.0f; B.y = lo ? w1_1 : 0.0f;
    v8f C;
#pragma unroll
    for (int r = 0; r < 8; ++r) C[r] = bias;

    float sc = 0.0f;
#pragma unroll
    for (int t = 0; t < 2; ++t) {
        // gather tile-t edge tangent vectors into lanes 0..15 (rows M=0..15)
        float axs = __shfl(vx, (lane & 15) + t * 16);
        float ays = __shfl(vy, (lane & 15) + t * 16);
        v2f A; A.x = lo ? axs : 0.0f; A.y = lo ? ays : 0.0f;

        // D[m][n] = vx*W1[0][n] + vy*W1[1][n] + b1[n]
        v8f D = __builtin_amdgcn_wmma_f32_16x16x4_f32(
            false, A, false, B, (short)0, C, false, false);

        // gelu, *W2[n], reduce across the 16 columns of each half-wave
        float p[8];
#pragma unroll
        for (int r = 0; r < 8; ++r) {
            float pr = gelu_exact(D[r]) * w2v;
            pr += __shfl_xor(pr, 1, 16);
            pr += __shfl_xor(pr, 2, 16);
            pr += __shfl_xor(pr, 4, 16);
            pr += __shfl_xor(pr, 8, 16);
            p[r] = pr;   // lanes 0..15 hold score[m=r]; lanes 16..31 score[m=8+r]
        }
        // each lane selects p[lane&7]; edge m reads it from lane m + (m&8)
        int r7 = lane & 7;
        float pick = p[0];
#pragma unroll
        for (int i = 1; i < 8; ++i) pick = (r7 == i) ? p[i] : pick;
        int m = lane & 15;
        float s = __shfl(pick, m + (m & 8));
        if ((lane >> 4) == t) sc = s;   // lanes of tile t keep their edge score
    }
    return sc;
}

__global__ void k_init(unsigned* __restrict__ mxkey, float* __restrict__ den,
                       float* __restrict__ msum) {
    int v = blockIdx.x * blockDim.x + threadIdx.x;
    mxkey[v]       = 0x007FFFFFu;   // fkey(-inf)
    den[v]         = 0.0f;
    msum[2 * v]     = 0.0f;
    msum[2 * v + 1] = 0.0f;
}

__global__ void k_edge(const float* __restrict__ x, const int* __restrict__ ei,
                       const float* __restrict__ W1, const float* __restrict__ b1,
                       const float* __restrict__ W2,
                       unsigned* __restrict__ mxkey, float* __restrict__ den,
                       float* __restrict__ msum, int pass) {
    const int e    = blockIdx.x * blockDim.x + threadIdx.x;   // E % 256 == 0
    const int lane = threadIdx.x & 31;
    const int nidx = lane & 15;

    const int row = ei[e];
    const int col = ei[E + e];
    const float xix = x[2 * row], xiy = x[2 * row + 1];
    const float xjx = x[2 * col], xjy = x[2 * col + 1];

    // v_ij = log_{xi}(xj):  sub = mobius_add(-xi, xj)
    const float ax = -xix, ay = -xiy;
    const float a2 = ax * ax + ay * ay;
    const float b2 = xjx * xjx + xjy * xjy;
    const float ab = ax * xjx + ay * xjy;
    const float coA  = 1.0f + 2.0f * ab + b2;
    const float coB  = 1.0f - a2;
    const float denm = fmaxf(1.0f + 2.0f * ab + a2 * b2, EPSF);
    const float sx = (coA * ax + coB * xjx) / denm;
    const float sy = (coA * ay + coB * xjy) / denm;
    const float nn = fmaxf(sqrtf(sx * sx + sy * sy), EPSF);
    const float fac = fmaxf(1.0f - (xix * xix + xiy * xiy), EPSF) *
                      atanhf(fminf(nn, MAX_ATANH)) / nn;
    const float vx = fac * sx, vy = fac * sy;

    // MLP scorer via WMMA (uniform control flow -> EXEC all ones)
    const float w1_0 = W1[nidx];
    const float w1_1 = W1[16 + nidx];
    const float bias = b1[nidx];
    const float w2v  = W2[nidx];
    const float sc = wave_mlp_score(vx, vy, w1_0, w1_1, bias, w2v);

    if (pass == 0) {
        atomicMax(&mxkey[row], fkey(sc));
    } else {
        const float mx = fdec(mxkey[row]);
        const float ee = expf(sc - mx);
        atomicAdd(&den[row], ee);
        atomicAdd(&msum[2 * row],     ee * vx);
        atomicAdd(&msum[2 * row + 1], ee * vy);
    }
}

__global__ void k_node(const float* __restrict__ x, const float* __restrict__ den,
                       const float* __restrict__ msum, const float* __restrict__ etap,
                       const float* __restrict__ dscale, const float* __restrict__ dtheta,
                       const int* __restrict__ depth, float* __restrict__ out) {
    const int v = blockIdx.x * blockDim.x + threadIdx.x;

    const float dn = fmaxf(den[v], EPSF);
    float m0 = msum[2 * v]     / dn;
    float m1 = msum[2 * v + 1] / dn;

    int d = depth[v];
    d = d < 0 ? 0 : (d > 511 ? 511 : d);
    const float k  = dscale[d];
    const float an = dtheta[d];
    const float ca = cosf(an), sa = sinf(an);
    const float r0 = k * (ca * m0 - sa * m1);
    const float r1 = k * (sa * m0 + ca * m1);

    const float eta = etap[0];
    const float ux = eta * r0, uy = eta * r1;

    const float xx = x[2 * v], xy = x[2 * v + 1];
    const float x2 = xx * xx + xy * xy;
    const float nn = fmaxf(sqrtf(ux * ux + uy * uy), EPSF);
    const float th = tanhf(nn / fmaxf(1.0f - x2, EPSF));   // tanh(0.5*lam*n)
    const float gx = th * ux / nn, gy = th * uy / nn;

    // mobius_add(x, g)
    const float b2 = gx * gx + gy * gy;
    const float ab = xx * gx + xy * gy;
    const float coA = 1.0f + 2.0f * ab + b2;
    const float coB = 1.0f - x2;
    const float dd  = fmaxf(1.0f + 2.0f * ab + x2 * b2, EPSF);
    out[2 * v]     = (coA * xx + coB * gx) / dd;
    out[2 * v + 1] = (coA * xy + coB * gy) / dd;
}

extern "C" void kernel_launch(void* const* d_in, const int* in_sizes, int n_in,
                              void* d_out, int out_size, void* d_ws, size_t ws_size,
                              hipStream_t stream) {
    const float* x      = (const float*)d_in[0];   // [V,2]
    const float* W1     = (const float*)d_in[1];   // [2,16]
    const float* b1     = (const float*)d_in[2];   // [16]
    const float* W2     = (const float*)d_in[3];   // [16,1]
    const float* eta    = (const float*)d_in[4];   // [1]
    const float* dscale = (const float*)d_in[5];   // [512]
    const float* dtheta = (const float*)d_in[6];   // [512]
    const int*   ei     = (const int*)d_in[7];     // [2,E]
    const int*   depth  = (const int*)d_in[8];     // [V]
    float*       out    = (float*)d_out;           // [V,2]

    // workspace: uint mxkey[V] | float den[V] | float msum[2V]  (~33.5 MB)
    unsigned* mxkey = (unsigned*)d_ws;
    float*    den   = (float*)(mxkey + V);
    float*    msum  = den + V;

    const int TB = 256;
    k_init<<<V / TB, TB, 0, stream>>>(mxkey, den, msum);
    k_edge<<<E / TB, TB, 0, stream>>>(x, ei, W1, b1, W2, mxkey, den, msum, 0);
    k_edge<<<E / TB, TB, 0, stream>>>(x, ei, W1, b1, W2, mxkey, den, msum, 1);
    k_node<<<V / TB, TB, 0, stream>>>(x, den, msum, eta, dscale, dtheta, depth, out);
}